// ConvTeNet_12979391169074
// MI455X (gfx1250) — compile-verified
//
#include <hip/hip_runtime.h>

typedef float v2f __attribute__((ext_vector_type(2)));
typedef float v8f __attribute__((ext_vector_type(8)));

#define D 20
#define BATCH 64

__device__ __forceinline__ v8f wmma_16x16x4_f32(v2f a, v2f b, v8f c) {
    // 8 args: (neg_a, A, neg_b, B, c_mod, C, reuse_a, reuse_b); f32 has no A/B neg -> false
    return __builtin_amdgcn_wmma_f32_16x16x4_f32(false, a, false, b, (short)0, c, false, false);
}

// One block per patch p. Computes y[b,p,:] = MPS(patch) then per-patch BatchNorm,
// writing dst flat as [B][P][D] (which later stages reinterpret with the raw-reshape rule).
__global__ __launch_bounds__(256) void mps_stage_kernel(
    const float* __restrict__ src,    // [B, C, H, W] flat (prev stage raw view)
    const float* __restrict__ cores,  // [P, S, 8, 400]
    const float* __restrict__ label,  // [20,20,20]
    const float* __restrict__ alpha,
    const float* __restrict__ omega,
    const float* __restrict__ gamma,
    const float* __restrict__ beta,
    float* __restrict__ dst,          // [B, P, 20]
    int P, int S, int C, int H, int W)
{
    __shared__ float sPhi[BATCH][8];
    __shared__ float sV [BATCH][D];
    __shared__ float sVn[BATCH][D];
    __shared__ float sW2[D * D];
    __shared__ float sRed[2];

    const int p    = blockIdx.x;
    const int t    = threadIdx.x;
    const int lane = t & 31;
    const int wave = t >> 5;            // 8 wave32 per block
    const int HW   = H * W;
    const int CHW  = C * HW;
    const int PC   = P * C;

    // v = alpha (identity start of scan folded in), vnew = 0
    for (int idx = t; idx < BATCH * D; idx += 256) {
        sV [idx / D][idx % D] = alpha[idx % D];
        sVn[idx / D][idx % D] = 0.0f;
    }
    // W2[j][o] = sum_k label[o][j][k] * omega[k]
    for (int idx = t; idx < D * D; idx += 256) {
        int j = idx / D, o = idx % D;
        float acc = 0.0f;
        for (int k = 0; k < D; ++k) acc += label[(o * D + j) * D + k] * omega[k];
        sW2[j * D + o] = acc;
    }
    if (t == 0) { sRed[0] = 0.0f; sRed[1] = 0.0f; }
    __syncthreads();

    for (int s = 0; s < S; ++s) {
        // ---- load phi for this (p, s): faithful raw-reshape unfold gather ----
        {
            int b = t >> 2, kk = t & 3;            // 64*4 = 256 loads
            int f  = kk * PC + p * C + s;          // flat idx into [4, P, C]
            int c  = f / HW;   int r1 = f - c * HW;    // decompose over [C, H/2, W/2, 2, 2]
            int W2x = W * 2;
            int h2 = r1 / W2x; int r2 = r1 - h2 * W2x;
            int w2 = r2 >> 2;  int r3 = r2 & 3;
            int ii = r3 >> 1,  jj = r3 & 1;
            float val = src[(long)b * CHW + c * HW + (h2 * 2 + ii) * W + (w2 * 2 + jj)];
            sPhi[b][kk]     = val;
            sPhi[b][kk + 4] = 1.0f - val;
        }
        __syncthreads();

        const float* cs = cores + (long)(p * S + s) * (8 * 400);
        if (s + 1 < S) __builtin_prefetch(cs + 8 * 400, 0, 1);   // global_prefetch next site

        // ---- M_all = Phi[64x8] x C_s[8x400] via v_wmma_f32_16x16x4, fused v-update ----
        for (int tile = wave; tile < 100; tile += 8) {
            int bt = tile / 25, ct = tile - bt * 25;
            int mrow = bt * 16 + (lane & 15);        // A: M = lane%16
            int aoff = (lane >> 4) << 1;             // A: K base 0 (lo half) / 2 (hi half)
            int coln = ct * 16 + (lane & 15);        // B/D: N = lane%16
            v8f acc = {};
            #pragma unroll
            for (int ks = 0; ks < 2; ++ks) {         // K = 8 -> 2 steps of 4
                v2f a; a[0] = sPhi[mrow][ks * 4 + aoff];
                       a[1] = sPhi[mrow][ks * 4 + aoff + 1];
                int kr = ks * 4 + aoff;
                v2f bb; bb[0] = cs[kr * 400 + coln];
                        bb[1] = cs[(kr + 1) * 400 + coln];
                acc = wmma_16x16x4_f32(a, bb, acc);
            }
            // D-tile: VGPR r -> row r (lanes 0-15) / r+8 (lanes 16-31)
            int mbase = bt * 16 + ((lane >> 4) << 3);
            int i = coln / D, j = coln - i * D;      // col = i*20 + j of the 20x20 matrix
            // Pre-gather the 8 v-values so the ds_loads clause together (one wait),
            // then issue 8 independent mul + ds_add_f32.
            float vi[8];
            #pragma unroll
            for (int r = 0; r < 8; ++r) vi[r] = sV[mbase + r][i];
            #pragma unroll
            for (int r = 0; r < 8; ++r) {
                atomicAdd(&sVn[mbase + r][j], vi[r] * acc[r]);   // ds_add_f32
            }
        }
        __syncthreads();
        for (int idx = t; idx < BATCH * D; idx += 256) {
            int b = idx / D, j = idx - (idx / D) * D;
            sV[b][j]  = sVn[b][j];
            sVn[b][j] = 0.0f;
        }
        __syncthreads();
    }

    // y[b][o] = sum_j v[b][j] * W2[j][o]   (store into sVn)
    for (int idx = t; idx < BATCH * D; idx += 256) {
        int b = idx / D, o = idx - (idx / D) * D;
        float acc = 0.0f;
        for (int j = 0; j < D; ++j) acc += sV[b][j] * sW2[j * D + o];
        sVn[b][o] = acc;
    }
    __syncthreads();

    // BatchNorm over (b, o) for this patch: biased var, eps=1e-5
    float ls = 0.0f, lq = 0.0f;
    for (int idx = t; idx < BATCH * D; idx += 256) {
        float y = sVn[idx / D][idx % D];
        ls += y; lq += y * y;
    }
    atomicAdd(&sRed[0], ls);
    atomicAdd(&sRed[1], lq);
    __syncthreads();
    const float invN = 1.0f / (float)(BATCH * D);
    float mu   = sRed[0] * invN;
    float var  = sRed[1] * invN - mu * mu;
    float rstd = rsqrtf(var + 1e-5f);
    float g = gamma[p], bofs = beta[p];
    for (int idx = t; idx < BATCH * D; idx += 256) {
        int b = idx / D, o = idx % D;
        dst[(long)b * (P * D) + p * D + o] = g * (sVn[b][o] - mu) * rstd + bofs;
    }
}

// Final MPS: P=1, S=64 sites, F=40, output [64,10]; no BN.
__global__ __launch_bounds__(256) void mps_final_kernel(
    const float* __restrict__ src,    // [64, 64, 20] = BN'd stage-3 output
    const float* __restrict__ cores,  // [1, 64, 40, 400]
    const float* __restrict__ label,  // [10, 20, 20]
    const float* __restrict__ alpha,
    const float* __restrict__ omega,
    float* __restrict__ out)          // [64, 10]
{
    __shared__ float sPhi[BATCH][40];
    __shared__ float sV [BATCH][D];
    __shared__ float sVn[BATCH][D];
    __shared__ float sW2[D * 10];

    const int t = threadIdx.x;
    const int lane = t & 31, wave = t >> 5;

    for (int idx = t; idx < BATCH * D; idx += 256) {
        sV [idx / D][idx % D] = alpha[idx % D];
        sVn[idx / D][idx % D] = 0.0f;
    }
    for (int idx = t; idx < D * 10; idx += 256) {
        int j = idx / 10, o = idx % 10;
        float acc = 0.0f;
        for (int k = 0; k < D; ++k) acc += label[(o * D + j) * D + k] * omega[k];
        sW2[j * 10 + o] = acc;
    }
    __syncthreads();

    for (int s = 0; s < 64; ++s) {
        for (int idx = t; idx < BATCH * D; idx += 256) {
            int b = idx / D, f = idx % D;
            float val = src[(long)b * (64 * D) + s * D + f];   // transpose: u[b,f,0,s]=y3[b,s,f]
            sPhi[b][f]      = val;
            sPhi[b][f + 20] = 1.0f - val;
        }
        __syncthreads();

        const float* cs = cores + (long)s * (40 * 400);
        if (s + 1 < 64) __builtin_prefetch(cs + 40 * 400, 0, 1);

        for (int tile = wave; tile < 100; tile += 8) {
            int bt = tile / 25, ct = tile - bt * 25;
            int mrow = bt * 16 + (lane & 15);
            int aoff = (lane >> 4) << 1;
            int coln = ct * 16 + (lane & 15);
            v8f acc = {};
            #pragma unroll
            for (int ks = 0; ks < 10; ++ks) {        // K = 40 -> 10 steps of 4
                v2f a; a[0] = sPhi[mrow][ks * 4 + aoff];
                       a[1] = sPhi[mrow][ks * 4 + aoff + 1];
                int kr = ks * 4 + aoff;
                v2f bb; bb[0] = cs[kr * 400 + coln];
                        bb[1] = cs[(kr + 1) * 400 + coln];
                acc = wmma_16x16x4_f32(a, bb, acc);
            }
            int mbase = bt * 16 + ((lane >> 4) << 3);
            int i = coln / D, j = coln - i * D;
            float vi[8];
            #pragma unroll
            for (int r = 0; r < 8; ++r) vi[r] = sV[mbase + r][i];
            #pragma unroll
            for (int r = 0; r < 8; ++r) {
                atomicAdd(&sVn[mbase + r][j], vi[r] * acc[r]);
            }
        }
        __syncthreads();
        for (int idx = t; idx < BATCH * D; idx += 256) {
            int b = idx / D, j = idx - (idx / D) * D;
            sV[b][j]  = sVn[b][j];
            sVn[b][j] = 0.0f;
        }
        __syncthreads();
    }

    for (int idx = t; idx < BATCH * 10; idx += 256) {
        int b = idx / 10, o = idx % 10;
        float acc = 0.0f;
        for (int j = 0; j < D; ++j) acc += sV[b][j] * sW2[j * 10 + o];
        out[b * 10 + o] = acc;
    }
}

extern "C" void kernel_launch(void* const* d_in, const int* in_sizes, int n_in,
                              void* d_out, int out_size, void* d_ws, size_t ws_size,
                              hipStream_t stream) {
    const float* x      = (const float*)d_in[0];
    const float* cores1 = (const float*)d_in[1];
    const float* label1 = (const float*)d_in[2];
    const float* alpha1 = (const float*)d_in[3];
    const float* omega1 = (const float*)d_in[4];
    const float* g1     = (const float*)d_in[5];
    const float* b1     = (const float*)d_in[6];
    const float* cores2 = (const float*)d_in[7];
    const float* label2 = (const float*)d_in[8];
    const float* alpha2 = (const float*)d_in[9];
    const float* omega2 = (const float*)d_in[10];
    const float* g2     = (const float*)d_in[11];
    const float* b2     = (const float*)d_in[12];
    const float* cores3 = (const float*)d_in[13];
    const float* label3 = (const float*)d_in[14];
    const float* alpha3 = (const float*)d_in[15];
    const float* omega3 = (const float*)d_in[16];
    const float* g3     = (const float*)d_in[17];
    const float* b3     = (const float*)d_in[18];
    const float* coresF = (const float*)d_in[19];
    const float* labelF = (const float*)d_in[20];
    const float* alphaF = (const float*)d_in[21];
    const float* omegaF = (const float*)d_in[22];

    float* ws = (float*)d_ws;
    float* y1 = ws;                               // 64*1024*20 = 1,310,720 f
    float* y2 = y1 + (size_t)64 * 1024 * 20;      // 64*256*20  =   327,680 f
    float* y3 = y2 + (size_t)64 * 256 * 20;       // 64*64*20   =    81,920 f

    mps_stage_kernel<<<1024, 256, 0, stream>>>(x,  cores1, label1, alpha1, omega1, g1, b1, y1,
                                               1024, 3, 3, 64, 64);
    mps_stage_kernel<<<256, 256, 0, stream>>>(y1, cores2, label2, alpha2, omega2, g2, b2, y2,
                                              256, 20, 20, 32, 32);
    mps_stage_kernel<<<64, 256, 0, stream>>>(y2, cores3, label3, alpha3, omega3, g3, b3, y3,
                                             64, 20, 20, 16, 16);
    mps_final_kernel<<<1, 256, 0, stream>>>(y3, coresF, labelF, alphaF, omegaF, (float*)d_out);
}